// BatchODE_45646912421934
// MI455X (gfx1250) — compile-verified
//
#include <hip/hip_runtime.h>
#include <hip/hip_bf16.h>

typedef float v2f __attribute__((ext_vector_type(2)));
typedef float v8f __attribute__((ext_vector_type(8)));

#define K_DIM 3
#define H_DIM 64
#define IN_DIM 7

// One wave (32 lanes) per batch element. Single-wave workgroup -> barriers are NOPs.
__global__ __launch_bounds__(32) void batch_ode_wmma_kernel(
    const float* __restrict__ state,      // (B, 6)
    const float* __restrict__ tscalar,    // (1,)
    const float* __restrict__ w1,         // (B, 64, 7)
    const float* __restrict__ b1,         // (B, 64)
    const float* __restrict__ w2,         // (B, 64, 64)
    const float* __restrict__ b2,         // (B, 64)
    const float* __restrict__ w3,         // (B, 3, 64)
    const float* __restrict__ b3,         // (B, 3)
    const float* __restrict__ log_omega,  // (B, 3)
    const float* __restrict__ log_gamma,  // (B, 3)
    float* __restrict__ out,              // (B, 6)
    int B)
{
    __shared__ float lds_h1[H_DIM];
    __shared__ float lds_h2[H_DIM];

    const int b    = blockIdx.x;
    const int lane = threadIdx.x;
    if (b >= B) return;

    const float* __restrict__ w1b = w1 + (size_t)b * H_DIM * IN_DIM;
    const float* __restrict__ w2b = w2 + (size_t)b * H_DIM * H_DIM;
    const float* __restrict__ w3b = w3 + (size_t)b * K_DIM * H_DIM;

    // ---- build input x = [state(6), t] (broadcast loads, every lane keeps a copy)
    float x[IN_DIM];
#pragma unroll
    for (int j = 0; j < 6; ++j) x[j] = state[b * 6 + j];
    x[6] = tscalar[0];

    // ---- layer 1 (64x7 matvec): each lane computes rows {lane, lane+32}
#pragma unroll
    for (int r = 0; r < 2; ++r) {
        const int row = lane + r * 32;
        const float* __restrict__ wr = w1b + row * IN_DIM;
        float acc = b1[b * H_DIM + row];
#pragma unroll
        for (int j = 0; j < IN_DIM; ++j) acc = fmaf(wr[j], x[j], acc);
        lds_h1[row] = tanhf(acc);
    }
    __syncthreads();

    // ---- layer 2 (64x64 matvec) via V_WMMA_F32_16X16X4_F32
    // A tile = w2[i0 .. i0+15][g0 .. g0+3]  (documented 16x4 f32 A layout)
    // B tile = h1[g0 .. g0+3] replicated across all 16 columns
    const int rowSel = lane & 15;         // M within tile
    const int koff   = (lane >> 4) << 1;  // lanes 0-15 -> K{0,1}, lanes 16-31 -> K{2,3}

    v8f acc0 = {}, acc1 = {}, acc2 = {}, acc3 = {};

#pragma unroll 4
    for (int g0 = 0; g0 < H_DIM; g0 += 4) {
        v2f bv;
        bv.x = lds_h1[g0 + koff];
        bv.y = lds_h1[g0 + koff + 1];

        const float* __restrict__ base = w2b + g0 + koff;
        v2f a0 = *(const v2f*)(base + (0  + rowSel) * H_DIM);
        v2f a1 = *(const v2f*)(base + (16 + rowSel) * H_DIM);
        v2f a2 = *(const v2f*)(base + (32 + rowSel) * H_DIM);
        v2f a3 = *(const v2f*)(base + (48 + rowSel) * H_DIM);

        acc0 = __builtin_amdgcn_wmma_f32_16x16x4_f32(false, a0, false, bv, (short)0, acc0, false, false);
        acc1 = __builtin_amdgcn_wmma_f32_16x16x4_f32(false, a1, false, bv, (short)0, acc1, false, false);
        acc2 = __builtin_amdgcn_wmma_f32_16x16x4_f32(false, a2, false, bv, (short)0, acc2, false, false);
        acc3 = __builtin_amdgcn_wmma_f32_16x16x4_f32(false, a3, false, bv, (short)0, acc3, false, false);
    }

    // ---- extract D (all 16 columns identical). C/D layout: VGPR j, lanes 0-15 hold
    // row M=j, lanes 16-31 hold row M=8+j. Lanes 0 and 16 dump their column to LDS.
    if (lane == 0 || lane == 16) {
        const int base = (lane >> 4) * 8;  // 0 or 8
#pragma unroll
        for (int j = 0; j < 8; ++j) {
            lds_h2[0  + base + j] = acc0[j];
            lds_h2[16 + base + j] = acc1[j];
            lds_h2[32 + base + j] = acc2[j];
            lds_h2[48 + base + j] = acc3[j];
        }
    }
    __syncthreads();

    // ---- bias + tanh: each lane owns h2 elements {2*lane, 2*lane+1}
    const int i0 = 2 * lane;
    const float h2a = tanhf(lds_h2[i0]     + b2[b * H_DIM + i0]);
    const float h2b = tanhf(lds_h2[i0 + 1] + b2[b * H_DIM + i0 + 1]);

    // ---- layer 3 (3x64 matvec): per-lane partials + wave32 butterfly reduction
    float p0 = w3b[0 * H_DIM + i0] * h2a + w3b[0 * H_DIM + i0 + 1] * h2b;
    float p1 = w3b[1 * H_DIM + i0] * h2a + w3b[1 * H_DIM + i0 + 1] * h2b;
    float p2 = w3b[2 * H_DIM + i0] * h2a + w3b[2 * H_DIM + i0 + 1] * h2b;
#pragma unroll
    for (int m = 16; m >= 1; m >>= 1) {
        p0 += __shfl_xor(p0, m, 32);
        p1 += __shfl_xor(p1, m, 32);
        p2 += __shfl_xor(p2, m, 32);
    }

    // ---- ODE output: lane k < 3 writes (dz, dv) for its oscillator
    if (lane < K_DIM) {
        const float corr = (lane == 0 ? p0 : (lane == 1 ? p1 : p2)) + b3[b * K_DIM + lane];
        const float om = expf(log_omega[b * K_DIM + lane]);
        const float ga = expf(log_gamma[b * K_DIM + lane]);
        const float zk = state[b * 6 + 2 * lane];
        const float vk = state[b * 6 + 2 * lane + 1];
        out[b * 6 + 2 * lane]     = vk;
        out[b * 6 + 2 * lane + 1] = fmaf(-2.0f * ga, vk, fmaf(-om * om, zk, corr));
    }
}

extern "C" void kernel_launch(void* const* d_in, const int* in_sizes, int n_in,
                              void* d_out, int out_size, void* d_ws, size_t ws_size,
                              hipStream_t stream) {
    const float* state     = (const float*)d_in[0];
    const float* t         = (const float*)d_in[1];
    const float* w1        = (const float*)d_in[2];
    const float* b1        = (const float*)d_in[3];
    const float* w2        = (const float*)d_in[4];
    const float* b2        = (const float*)d_in[5];
    const float* w3        = (const float*)d_in[6];
    const float* b3        = (const float*)d_in[7];
    const float* log_omega = (const float*)d_in[8];
    const float* log_gamma = (const float*)d_in[9];
    float* out = (float*)d_out;

    const int B = in_sizes[0] / 6;  // state is (B, 2K) with K=3
    batch_ode_wmma_kernel<<<B, 32, 0, stream>>>(state, t, w1, b1, w2, b2, w3, b3,
                                                log_omega, log_gamma, out, B);
}